// HierarchicalAttention_24842090840352
// MI455X (gfx1250) — compile-verified
//
#include <hip/hip_runtime.h>
#include <hip/hip_bf16.h>
#include <math.h>

typedef float v2f __attribute__((ext_vector_type(2)));
typedef float v8f __attribute__((ext_vector_type(8)));

namespace {
constexpr int   kBatch  = 2;
constexpr int   kHeads  = 16;
constexpr int   kSeq    = 4096;
constexpr int   kDim    = 64;
constexpr int   kLocalW = 128;
constexpr float kScale  = 0.125f;            // 1/sqrt(64)
constexpr int   kBH     = kBatch * kHeads;   // 32
constexpr int   kC1     = kSeq / 8;          // 512 chunks, cs=8
constexpr int   kC2     = kSeq / 64;         // 64  chunks, cs=64
constexpr int   kC3     = kSeq / 512;        // 8   chunks, cs=512
}

// ---------------------------------------------------------------------------
// 8x mean-pool along sequence: dst[bh,c,d] = mean_{i<8} src[bh, 8c+i, d]
// ---------------------------------------------------------------------------
__global__ void hattn_pool8(const float* __restrict__ src,
                            float* __restrict__ dst, int n_src) {
  const int    n_dst = n_src >> 3;
  const size_t total = (size_t)kBH * n_dst * kDim;
  size_t idx = (size_t)blockIdx.x * blockDim.x + threadIdx.x;
  if (idx >= total) return;
  const int    d  = (int)(idx & (kDim - 1));
  const size_t t  = idx >> 6;
  const int    c  = (int)(t % n_dst);
  const size_t bh = t / n_dst;
  const float* p  = src + (bh * (size_t)n_src + (size_t)c * 8) * kDim + d;
  float s = 0.f;
#pragma unroll
  for (int i = 0; i < 8; ++i) s += p[(size_t)i * kDim];
  dst[idx] = s * 0.125f;
}

// ---------------------------------------------------------------------------
// One 16(query) x 16(key/chunk) tile: S = Q*K^T via WMMA f32 16x16x4,
// mask + decay + online softmax, then O += P*V via WMMA (P through LDS).
// cs==0 -> local-window phase; cs>0 -> compressed level with chunk size cs.
// ---------------------------------------------------------------------------
__device__ __forceinline__ void hattn_tile(
    const float* __restrict__ kp, const float* __restrict__ vp,
    int t0, int nrows, int cs, float gamma, int q0, int lane,
    float* __restrict__ ldsP, const v2f* qa,
    float m_run[8], float l_run[8], v8f oacc[4])
{
  const int half = lane >> 4;     // 0 | 1
  const int lm   = lane & 15;

  // -- B fragments (K^T): lane holds key-row n = t0+lm, elems k = 4kk+2h+{0,1}
  int nrow = t0 + lm;
  nrow = nrow < 0 ? 0 : (nrow >= nrows ? nrows - 1 : nrow);
  const float* kpl = kp + (size_t)nrow * kDim + 2 * half;
  v2f kb[16];
#pragma unroll
  for (int kk = 0; kk < 16; ++kk) kb[kk] = *(const v2f*)(kpl + 4 * kk);

  // -- S = Q * K^T, K dim = 64 -> 16 x v_wmma_f32_16x16x4_f32
  v8f s = {0.f, 0.f, 0.f, 0.f, 0.f, 0.f, 0.f, 0.f};
#pragma unroll
  for (int kk = 0; kk < 16; ++kk)
    s = __builtin_amdgcn_wmma_f32_16x16x4_f32(false, qa[kk], false, kb[kk],
                                              (short)0, s, false, false);

  // -- mask, decay, online softmax. C/D layout: elem(r) = row r+8*half, col lm.
  const int key = t0 + lm;
  float scl[8];
#pragma unroll
  for (int r = 0; r < 8; ++r) {
    const int qg = q0 + r + 8 * half;
    bool  valid;
    float dist;
    if (cs == 0) {  // local window
      valid = (key >= 0) && (key <= qg) && (key >= qg - (kLocalW - 1));
      dist  = (float)(qg - key);
    } else {        // compressed level: chunk_end <= max(q-127,0) & in range
      valid = (key < nrows) && ((key + 1) * cs <= qg - (kLocalW - 1));
      dist  = fabsf((float)qg - ((float)key * (float)cs + 0.5f * (float)cs));
    }
    float sv = valid ? (s[r] * kScale - gamma * dist) : -INFINITY;

    float rm = sv;                       // row max over the 16-lane half
    rm = fmaxf(rm, __shfl_xor(rm, 1));
    rm = fmaxf(rm, __shfl_xor(rm, 2));
    rm = fmaxf(rm, __shfl_xor(rm, 4));
    rm = fmaxf(rm, __shfl_xor(rm, 8));

    const float m_new = fmaxf(m_run[r], rm);
    const float msafe = fmaxf(m_new, -1.0e9f);                 // ref m_safe
    const float sc    = __expf(fmaxf(m_run[r], -1.0e9f) - msafe);
    m_run[r] = m_new;

    const float p = valid ? __expf(sv - msafe) : 0.f;
    float ps = p;                        // row sum
    ps += __shfl_xor(ps, 1);
    ps += __shfl_xor(ps, 2);
    ps += __shfl_xor(ps, 4);
    ps += __shfl_xor(ps, 8);
    l_run[r] = l_run[r] * sc + ps;
    scl[r]   = sc;

    ldsP[(r + 8 * half) * 18 + lm] = p;  // P row-major, padded stride 18
  }

#pragma unroll
  for (int nc = 0; nc < 4; ++nc)
#pragma unroll
    for (int r = 0; r < 8; ++r) oacc[nc][r] *= scl[r];

  // -- O += P * V : A = P (16q x 16key, from LDS), B = V (16key x 16d chunk)
#pragma unroll
  for (int nc = 0; nc < 4; ++nc) {
#pragma unroll
    for (int kk = 0; kk < 4; ++kk) {
      const v2f pa = *(const v2f*)(ldsP + lm * 18 + kk * 4 + 2 * half);
      const int kr = t0 + kk * 4 + 2 * half;
      int kr0 = kr,     kr1 = kr + 1;
      kr0 = kr0 < 0 ? 0 : (kr0 >= nrows ? nrows - 1 : kr0);
      kr1 = kr1 < 0 ? 0 : (kr1 >= nrows ? nrows - 1 : kr1);
      v2f vb;
      vb.x = vp[(size_t)kr0 * kDim + nc * 16 + lm];
      vb.y = vp[(size_t)kr1 * kDim + nc * 16 + lm];
      oacc[nc] = __builtin_amdgcn_wmma_f32_16x16x4_f32(
          false, pa, false, vb, (short)0, oacc[nc], false, false);
    }
  }
}

// ---------------------------------------------------------------------------
// Main fused kernel: 1 wave == one 16-query tile of one (b,h).
// ---------------------------------------------------------------------------
__launch_bounds__(128)
__global__ void hattn_main(const float* __restrict__ q,
                           const float* __restrict__ k,
                           const float* __restrict__ v,
                           const float* __restrict__ k1, const float* __restrict__ v1,
                           const float* __restrict__ k2, const float* __restrict__ v2,
                           const float* __restrict__ k3, const float* __restrict__ v3,
                           const float* __restrict__ gammas,
                           float* __restrict__ out)
{
  __shared__ float ldsP[4][16 * 18];

  const int lane = threadIdx.x & 31;
  const int wid  = threadIdx.x >> 5;
  const int gw   = blockIdx.x * 4 + wid;
  const int nQT  = kSeq / 16;                 // 256 query tiles per head
  const int bh   = gw / nQT;
  const int qt   = gw - bh * nQT;
  const int q0   = qt * 16;
  const size_t base = (size_t)bh * kSeq * kDim;
  const int half = lane >> 4, lm = lane & 15;
  float* lp = &ldsP[wid][0];

  // A fragments of the whole Q tile: lane=row q0+lm, elems k = 4kk+2h+{0,1}
  v2f qa[16];
  const float* qp = q + base + (size_t)(q0 + lm) * kDim + 2 * half;
#pragma unroll
  for (int kk = 0; kk < 16; ++kk) qa[kk] = *(const v2f*)(qp + 4 * kk);

  float m_run[8], l_run[8];
  v8f   oacc[4];
  const v8f vzero = {0.f, 0.f, 0.f, 0.f, 0.f, 0.f, 0.f, 0.f};
#pragma unroll
  for (int r = 0; r < 8; ++r) { m_run[r] = -INFINITY; l_run[r] = 0.f; }
#pragma unroll
  for (int nc = 0; nc < 4; ++nc) oacc[nc] = vzero;

  const float g0 = gammas[0], g1 = gammas[1], g2 = gammas[2], g3 = gammas[3];

  // ---- phase 0: local sliding window (keys [q-127, q]) ------------------
  {
    int lo = q0 - (kLocalW - 1);
    if (lo < 0) lo = 0;
    lo &= ~15;
    const float* kh = k + base;
    const float* vh = v + base;
    for (int t0 = lo; t0 <= q0; t0 += 16)
      hattn_tile(kh, vh, t0, kSeq, 0, g0, q0, lane, lp, qa, m_run, l_run, oacc);
  }

  // ---- phases 1..3: compressed levels -----------------------------------
  {
    const int hi = q0 + 15 - (kLocalW - 1);   // largest usable chunk_end
    const float* k1h = k1 + (size_t)bh * kC1 * kDim;
    const float* v1h = v1 + (size_t)bh * kC1 * kDim;
    const float* k2h = k2 + (size_t)bh * kC2 * kDim;
    const float* v2h = v2 + (size_t)bh * kC2 * kDim;
    const float* k3h = k3 + (size_t)bh * kC3 * kDim;
    const float* v3h = v3 + (size_t)bh * kC3 * kDim;

    int cmax = (hi >= 8) ? (hi / 8 - 1) : -1;           // level 1, cs = 8
    for (int t0 = 0; t0 <= cmax; t0 += 16)
      hattn_tile(k1h, v1h, t0, kC1, 8, g1, q0, lane, lp, qa, m_run, l_run, oacc);

    cmax = (hi >= 64) ? (hi / 64 - 1) : -1;             // level 2, cs = 64
    for (int t0 = 0; t0 <= cmax; t0 += 16)
      hattn_tile(k2h, v2h, t0, kC2, 64, g2, q0, lane, lp, qa, m_run, l_run, oacc);

    cmax = (hi >= 512) ? (hi / 512 - 1) : -1;           // level 3, cs = 512
    for (int t0 = 0; t0 <= cmax; t0 += 16)
      hattn_tile(k3h, v3h, t0, kC3, 512, g3, q0, lane, lp, qa, m_run, l_run, oacc);
  }

  // ---- epilogue: out = o / clip(l, 1e-8) --------------------------------
#pragma unroll
  for (int r = 0; r < 8; ++r) {
    const int   qg   = q0 + r + 8 * half;
    const float linv = 1.f / fmaxf(l_run[r], 1e-8f);
#pragma unroll
    for (int nc = 0; nc < 4; ++nc)
      out[base + (size_t)qg * kDim + nc * 16 + lm] = oacc[nc][r] * linv;
  }
}

// ---------------------------------------------------------------------------
extern "C" void kernel_launch(void* const* d_in, const int* in_sizes, int n_in,
                              void* d_out, int out_size, void* d_ws, size_t ws_size,
                              hipStream_t stream) {
  const float* q   = (const float*)d_in[0];
  const float* k   = (const float*)d_in[1];
  const float* v   = (const float*)d_in[2];
  const float* gam = (const float*)d_in[3];
  float* out = (float*)d_out;
  float* ws  = (float*)d_ws;

  const size_t n1 = (size_t)kBH * kC1 * kDim;   // 1,048,576 floats
  const size_t n2 = (size_t)kBH * kC2 * kDim;   //   131,072 floats
  const size_t n3 = (size_t)kBH * kC3 * kDim;   //    16,384 floats
  float* k1 = ws;        float* v1 = k1 + n1;
  float* k2 = v1 + n1;   float* v2 = k2 + n2;
  float* k3 = v2 + n2;   float* v3 = k3 + n3;   // total ~9.2 MB of d_ws

  auto gb = [](size_t n) { return dim3((unsigned)((n + 255) / 256)); };
  hattn_pool8<<<gb(n1), 256, 0, stream>>>(k,  k1, kSeq);
  hattn_pool8<<<gb(n1), 256, 0, stream>>>(v,  v1, kSeq);
  hattn_pool8<<<gb(n2), 256, 0, stream>>>(k1, k2, kC1);
  hattn_pool8<<<gb(n2), 256, 0, stream>>>(v1, v2, kC1);
  hattn_pool8<<<gb(n3), 256, 0, stream>>>(k2, k3, kC2);
  hattn_pool8<<<gb(n3), 256, 0, stream>>>(v2, v3, kC2);

  const int waves = kBH * (kSeq / 16);          // 8192 waves, 4 per block
  hattn_main<<<dim3(waves / 4), 128, 0, stream>>>(
      q, k, v, k1, v1, k2, v2, k3, v3, gam, out);
}